// GRU_base_50611894616115
// MI455X (gfx1250) — compile-verified
//
#include <hip/hip_runtime.h>
#include <math.h>

// Problem constants from the reference
#define T_STEPS 100
#define H_DIM   50
#define Z_DIM   10
#define OUT_DIM 65          // Z + Z*(Z+1)/2
#define LN_EPS  1e-6f

// Padded GEMM geometry (tiled for 16x16 WMMA, K padded to multiple of 4)
#define M_PAD   112         // 7 * 16  (>= T_STEPS)
#define K_PAD   52          // 13 * 4  (>= H_DIM)
#define N_PAD   80          // 5 * 16  (>= OUT_DIM)

typedef __attribute__((ext_vector_type(2))) float v2f;
typedef __attribute__((ext_vector_type(8))) float v8f;

// ---------------------------------------------------------------------------
// Kernel 1: sequential GRU scan. One workgroup (256 threads = 8 waves),
// weights staged in LDS (30 KB << 320 KB/WGP), h kept in LDS.
// The scan is a serial dependency chain -> latency bound; VALU matvec
// (150 output lanes x 50 fmacs) beats WMMA for M=1.
// Also pre-stages the padded GEMM operands for kernel 2 so that kernel 2's
// inner loop needs no bounds checks at all.
// ---------------------------------------------------------------------------
__global__ __launch_bounds__(256) void gru_scan_kernel(
    const float* __restrict__ carry_init,
    const float* __restrict__ W_hr, const float* __restrict__ b_hr,
    const float* __restrict__ s_r,  const float* __restrict__ o_r,
    const float* __restrict__ W_hz, const float* __restrict__ b_hz,
    const float* __restrict__ s_z,  const float* __restrict__ o_z,
    const float* __restrict__ W_hn, const float* __restrict__ b_hn,
    const float* __restrict__ s_n,  const float* __restrict__ o_n,
    const float* __restrict__ W_dense,  // [H][OUT_DIM]
    const float* __restrict__ b_dense,  // [OUT_DIM]
    float* __restrict__ hs_p,           // [M_PAD][K_PAD] zero-padded out
    float* __restrict__ WpT,            // [N_PAD][K_PAD] = padded W_dense^T
    float* __restrict__ b_pad)          // [N_PAD]
{
    __shared__ float Wl[3][H_DIM * H_DIM];        // row-major: W[k*H + j]
    __shared__ float bl[3][H_DIM], sl[3][H_DIM], ol[3][H_DIM];
    __shared__ float h[H_DIM];
    __shared__ float acc[3][H_DIM];               // pre-activations
    __shared__ float gate[3][H_DIM];              // r,z sigmoided; n = LN'd only
    __shared__ float mu_s[3], rs_s[3];

    const int tid = threadIdx.x;

    // --- zero-pad hs_p, build padded transposed W_dense and padded bias ---
    for (int i = tid; i < M_PAD * K_PAD; i += 256) hs_p[i] = 0.0f;
    for (int i = tid; i < N_PAD * K_PAD; i += 256) {
        const int n = i / K_PAD;
        const int k = i % K_PAD;
        WpT[i] = (n < OUT_DIM && k < H_DIM) ? W_dense[k * OUT_DIM + n] : 0.0f;
    }
    for (int i = tid; i < N_PAD; i += 256)
        b_pad[i] = (i < OUT_DIM) ? b_dense[i] : 0.0f;

    // --- stage recurrent weights into LDS ---
    for (int i = tid; i < H_DIM * H_DIM; i += 256) {
        Wl[0][i] = W_hr[i];
        Wl[1][i] = W_hz[i];
        Wl[2][i] = W_hn[i];
    }
    for (int i = tid; i < H_DIM; i += 256) {
        bl[0][i] = b_hr[i]; bl[1][i] = b_hz[i]; bl[2][i] = b_hn[i];
        sl[0][i] = s_r[i];  sl[1][i] = s_z[i];  sl[2][i] = s_n[i];
        ol[0][i] = o_r[i];  ol[1][i] = o_z[i];  ol[2][i] = o_n[i];
        h[i] = carry_init[i];
    }
    __syncthreads();

    for (int t = 0; t < T_STEPS; ++t) {
        // --- 3 matvecs: acc[g][j] = b[g][j] + sum_k h[k] * W[g][k*H+j] ---
        if (tid < 3 * H_DIM) {
            const int g = tid / H_DIM;
            const int j = tid % H_DIM;
            const float* W = Wl[g];
            float sum = bl[g][j];
            #pragma unroll 5
            for (int k = 0; k < H_DIM; ++k)
                sum = fmaf(h[k], W[k * H_DIM + j], sum);
            acc[g][j] = sum;
        }
        __syncthreads();

        // --- per-gate layernorm stats (3 tiny serial reductions) ---
        if (tid < 3) {
            float s1 = 0.0f, s2 = 0.0f;
            for (int k = 0; k < H_DIM; ++k) {
                float x = acc[tid][k];
                s1 += x; s2 += x * x;
            }
            float m = s1 * (1.0f / H_DIM);
            float v = s2 * (1.0f / H_DIM) - m * m;
            mu_s[tid] = m;
            rs_s[tid] = rsqrtf(v + LN_EPS);
        }
        __syncthreads();

        // --- LN + activation ---
        if (tid < 3 * H_DIM) {
            const int g = tid / H_DIM;
            const int j = tid % H_DIM;
            float x = (acc[g][j] - mu_s[g]) * rs_s[g] * sl[g][j] + ol[g][j];
            gate[g][j] = (g < 2) ? (1.0f / (1.0f + expf(-x))) : x;
        }
        __syncthreads();

        // --- combine: new_h = (1-z)*tanh(r*LN_n) + z*h ---
        if (tid < H_DIM) {
            float r = gate[0][tid];
            float z = gate[1][tid];
            float n = tanhf(r * gate[2][tid]);
            float nh = (1.0f - z) * n + z * h[tid];
            h[tid] = nh;
            hs_p[t * K_PAD + tid] = nh;    // padded row, cols 50..51 stay 0
        }
        __syncthreads();
    }
}

// ---------------------------------------------------------------------------
// Kernel 2: out_p = hs_p(112x52) @ WpT^T(52x80) + b_pad, via full-f32 WMMA
// V_WMMA_F32_16X16X4_F32 (keeps reference f32 precision).
// One wave per 16x16 tile; 7 M-tiles x 5 N-tiles = 35 waves; 13 K-steps.
// Everything is pre-padded -> the loop is branch-free: each K-step is one
// global_load_b64 for A, one for B (both lane-contiguous), and one v_wmma.
// Fragment layouts per ISA 7.12.2:
//   A 16x4 : lane m = lane&15; K = kb + (lane>>4)*2 + v   (v = vgpr 0/1)
//   B 4x16 : lane n = lane&15; K = kb + (lane>>4)*2 + v
//   C/D    : 8 vgprs; M = (lane>>4)*8 + v; N = lane&15
// blockDim == 32, no divergence anywhere -> EXEC all ones (WMMA requirement).
// ---------------------------------------------------------------------------
__global__ __launch_bounds__(32) void dense_wmma_kernel(
    const float* __restrict__ hs_p,   // [M_PAD][K_PAD]
    const float* __restrict__ WpT,    // [N_PAD][K_PAD]
    const float* __restrict__ b_pad,  // [N_PAD]
    float* __restrict__ out_p)        // [M_PAD][N_PAD]
{
    const int lane = threadIdx.x;        // 0..31
    const int tm = blockIdx.x / 5;       // 7 M tiles
    const int tn = blockIdx.x % 5;       // 5 N tiles

    const int mrow = tm * 16 + (lane & 15);
    const int ncol = tn * 16 + (lane & 15);
    const int khi  = (lane >> 4) * 2;

    const float* arow = hs_p + mrow * K_PAD + khi;   // 8B-aligned pairs
    const float* brow = WpT  + ncol * K_PAD + khi;

    v8f c = {};
    #pragma unroll
    for (int kb = 0; kb < K_PAD; kb += 4) {
        const v2f a = *(const v2f*)(arow + kb);
        const v2f b = *(const v2f*)(brow + kb);
        // 8 args: (neg_a, A, neg_b, B, c_mod, C, reuse_a, reuse_b)
        c = __builtin_amdgcn_wmma_f32_16x16x4_f32(
                false, a, false, b, (short)0, c, false, false);
    }

    const float bias = b_pad[ncol];
    #pragma unroll
    for (int v = 0; v < 8; ++v) {
        const int m = tm * 16 + (lane >> 4) * 8 + v;
        out_p[m * N_PAD + ncol] = c[v] + bias;
    }
}

// ---------------------------------------------------------------------------
// Kernel 3: per-timestep natural-parameter transform (10x10 Cholesky algebra).
// 100 independent ~2KFLOP tasks -> one thread per task.
//   L from tril(flat), diag <- softplus; Sigma = L L^T;
//   J = (L L^T)^{-1} via L Y = I then L^T J = Y; h_nat = J mu.
// ---------------------------------------------------------------------------
__global__ __launch_bounds__(32) void nat_kernel(
    const float* __restrict__ out_p,     // [M_PAD][N_PAD] (rows 0..T-1 valid)
    float* __restrict__ Sigma,           // [T][Z][Z]
    float* __restrict__ mu_o,            // [T][Z]
    float* __restrict__ Jo,              // [T][Z][Z]
    float* __restrict__ hn)              // [T][Z]
{
    const int t = blockIdx.x * 32 + threadIdx.x;
    if (t >= T_STEPS) return;
    const float* x = out_p + t * N_PAD;

    float L[Z_DIM][Z_DIM];
    for (int i = 0; i < Z_DIM; ++i)
        for (int j = 0; j < Z_DIM; ++j)
            L[i][j] = 0.0f;
    {
        int idx = 0;
        for (int i = 0; i < Z_DIM; ++i)
            for (int j = 0; j <= i; ++j)
                L[i][j] = x[Z_DIM + idx++];
    }
    for (int i = 0; i < Z_DIM; ++i) {
        float d = L[i][i];                       // stable softplus
        L[i][i] = log1pf(expf(-fabsf(d))) + fmaxf(d, 0.0f);
    }

    // Sigma = L L^T
    for (int i = 0; i < Z_DIM; ++i)
        for (int j = 0; j < Z_DIM; ++j) {
            float s = 0.0f;
            const int kmax = (i < j) ? i : j;
            for (int k = 0; k <= kmax; ++k) s = fmaf(L[i][k], L[j][k], s);
            Sigma[t * Z_DIM * Z_DIM + i * Z_DIM + j] = s;
        }

    // Y = L^{-1} (forward substitution, columns of I)
    float Y[Z_DIM][Z_DIM];
    for (int c = 0; c < Z_DIM; ++c)
        for (int i = 0; i < Z_DIM; ++i) {
            if (i < c) { Y[i][c] = 0.0f; continue; }
            float s = (i == c) ? 1.0f : 0.0f;
            for (int k = c; k < i; ++k) s -= L[i][k] * Y[k][c];
            Y[i][c] = s / L[i][i];
        }

    // Solve L^T J = Y (back substitution) -> J = (L L^T)^{-1}
    float Jm[Z_DIM][Z_DIM];
    for (int c = 0; c < Z_DIM; ++c)
        for (int i = Z_DIM - 1; i >= 0; --i) {
            float s = Y[i][c];
            for (int k = i + 1; k < Z_DIM; ++k) s -= L[k][i] * Jm[k][c];
            Jm[i][c] = s / L[i][i];
        }

    float m[Z_DIM];
    for (int i = 0; i < Z_DIM; ++i) m[i] = x[i];
    for (int i = 0; i < Z_DIM; ++i) {
        float s = 0.0f;
        for (int c = 0; c < Z_DIM; ++c) s = fmaf(Jm[i][c], m[c], s);
        hn[t * Z_DIM + i]   = s;
        mu_o[t * Z_DIM + i] = m[i];
        for (int c = 0; c < Z_DIM; ++c)
            Jo[t * Z_DIM * Z_DIM + i * Z_DIM + c] = Jm[i][c];
    }
}

// ---------------------------------------------------------------------------
extern "C" void kernel_launch(void* const* d_in, const int* in_sizes, int n_in,
                              void* d_out, int out_size, void* d_ws, size_t ws_size,
                              hipStream_t stream) {
    const float* carry   = (const float*)d_in[0];
    const float* W_hr    = (const float*)d_in[1];
    const float* b_hr    = (const float*)d_in[2];
    const float* s_r     = (const float*)d_in[3];
    const float* o_r     = (const float*)d_in[4];
    const float* W_hz    = (const float*)d_in[5];
    const float* b_hz    = (const float*)d_in[6];
    const float* s_z     = (const float*)d_in[7];
    const float* o_z     = (const float*)d_in[8];
    const float* W_hn    = (const float*)d_in[9];
    const float* b_hn    = (const float*)d_in[10];
    const float* s_n     = (const float*)d_in[11];
    const float* o_n     = (const float*)d_in[12];
    const float* W_dense = (const float*)d_in[13];
    const float* b_dense = (const float*)d_in[14];

    float* ws    = (float*)d_ws;
    float* hs_p  = ws;                                   // M_PAD*K_PAD = 5824
    float* WpT   = hs_p + M_PAD * K_PAD;                 // N_PAD*K_PAD = 4160
    float* b_pad = WpT + N_PAD * K_PAD;                  // N_PAD       =   80
    float* out_p = b_pad + N_PAD;                        // M_PAD*N_PAD = 8960

    float* o     = (float*)d_out;               // 22000 floats total
    float* Sigma = o;                                                   // 10000
    float* mu_o  = o + T_STEPS * Z_DIM * Z_DIM;                         //  1000
    float* Jo    = o + T_STEPS * Z_DIM * Z_DIM + T_STEPS * Z_DIM;       // 10000
    float* hn    = o + 2 * T_STEPS * Z_DIM * Z_DIM + T_STEPS * Z_DIM;   //  1000

    gru_scan_kernel<<<1, 256, 0, stream>>>(
        carry, W_hr, b_hr, s_r, o_r, W_hz, b_hz, s_z, o_z,
        W_hn, b_hn, s_n, o_n, W_dense, b_dense,
        hs_p, WpT, b_pad);

    dense_wmma_kernel<<<35, 32, 0, stream>>>(hs_p, WpT, b_pad, out_p);

    nat_kernel<<<(T_STEPS + 31) / 32, 32, 0, stream>>>(
        out_p, Sigma, mu_o, Jo, hn);
}